// MultiHeadAttentionSequentialBatched_19756849561622
// MI455X (gfx1250) — compile-verified
//
#include <hip/hip_runtime.h>

// Problem constants (match reference)
#define B_ 16
#define H_ 32
#define D_ 4096
#define K_ 128
#define V_ 128
#define M_ 2048

typedef __attribute__((ext_vector_type(2))) float v2f;
typedef __attribute__((ext_vector_type(4))) float v4f;
typedef __attribute__((ext_vector_type(8))) float v8f;

// ---------------------------------------------------------------------------
// Kernel 1: QKV projections via V_WMMA_F32_16X16X4_F32.
// One block (4 waves) per (tensor, head, 16-wide k tile). Each wave
// accumulates the 16(batch) x 16(k) tile over a 1024-wide slice of D
// (WMMA K=4 per step); partials combined through LDS; wave 0 scatters.
// f32 A layout: lane m=lane&15, VGPR0 = K {0|2}, VGPR1 = K {1|3} per half-wave.
// f32 B layout (symmetric): VGPR0 = rows {0|2}, VGPR1 = rows {1|3}, lane = N.
// ---------------------------------------------------------------------------
__global__ __launch_bounds__(128)
void qkv_proj_wmma(const float* __restrict__ x,
                   const float* __restrict__ Wq,
                   const float* __restrict__ Wk,
                   const float* __restrict__ Wv,
                   float* __restrict__ q_ws,
                   float* __restrict__ Kc,
                   float* __restrict__ Vc) {
  __shared__ float part[4][32][8];

  const int lane = threadIdx.x & 31;
  const int wave = threadIdx.x >> 5;       // 0..3 : slice of the D reduction
  int id = blockIdx.x;                     // 0 .. 3*H_*8-1
  const int which = id / (H_ * 8);         // 0=q, 1=k, 2=v
  id %= (H_ * 8);
  const int h  = id / 8;
  const int k0 = (id % 8) * 16;

  const float* W  = (which == 0) ? Wq : (which == 1) ? Wk : Wv;
  const float* Wh = W + (size_t)h * D_ * K_;

  const int n   = lane & 15;               // tile column (k)
  const int cof = (lane < 16) ? 0 : 2;     // d sub-offset for this half-wave

  v8f acc = {};
  const int dbeg = wave * (D_ / 4);
  const int dend = dbeg + (D_ / 4);
  for (int d0 = dbeg; d0 < dend; d0 += 4) {
    const float* xp = x + (size_t)n * D_ + d0 + cof;         // A: x[m=n][d0+cof..+1]
    v2f a; a.x = xp[0]; a.y = xp[1];
    const float* wp = Wh + (size_t)(d0 + cof) * K_ + k0 + n; // B: W[d0+cof..][k0+n]
    v2f bt; bt.x = wp[0]; bt.y = wp[K_];
    acc = __builtin_amdgcn_wmma_f32_16x16x4_f32(false, a, false, bt,
                                                (short)0, acc, false, false);
  }

  if (wave > 0) {
    #pragma unroll
    for (int r = 0; r < 8; ++r) part[wave][lane][r] = acc[r];
  }
  __syncthreads();
  if (wave == 0) {
    #pragma unroll
    for (int r = 0; r < 8; ++r)
      acc[r] += part[1][lane][r] + part[2][lane][r] + part[3][lane][r];

    // Branch-free scatter: pick destination base + batch stride once.
    float* dst;
    size_t stride;
    if (which == 0) {
      dst    = q_ws + (size_t)h * K_ + k0 + n;
      stride = (size_t)H_ * K_;
    } else {
      float* base = (which == 1) ? Kc : Vc;
      dst    = base + ((size_t)h * (M_ + 1) + M_) * K_ + k0 + n; // new row m=M_
      stride = (size_t)H_ * (M_ + 1) * K_;
    }
    const int mbase = (lane < 16) ? 0 : 8;
    #pragma unroll
    for (int r = 0; r < 8; ++r)
      dst[(size_t)(mbase + r) * stride] = acc[r];
  }
}

// ---------------------------------------------------------------------------
// Kernel 2: fused cache-append + single-query attention.
// One block (256 thr = 8 waves) per (b,h). Streams prev_K / prev_V exactly
// once from HBM with NT hints (read-once / write-once, never re-read),
// writing them through to Kc / Vc while computing logits and the weighted
// V sum — saves a full 1 GB re-read vs copy-then-read and keeps L2 free
// for the reused weights.
// ---------------------------------------------------------------------------
__global__ __launch_bounds__(256)
void attn_fused(const float* __restrict__ prevK,
                const float* __restrict__ prevV,
                const float* __restrict__ q_ws,
                float* __restrict__ Kc,
                float* __restrict__ Vc,
                float* __restrict__ o_ws) {
  __shared__ float qs[K_];
  __shared__ float logits[M_ + 1];
  __shared__ float red[256];
  __shared__ float s_inv;

  const int bh   = blockIdx.x;             // b*H + h
  const int tid  = threadIdx.x;
  const int lane = tid & 31;
  const int wave = tid >> 5;

  const float* pK = prevK + (size_t)bh * M_ * K_;
  const float* pV = prevV + (size_t)bh * M_ * V_;
  float* cK = Kc + (size_t)bh * (M_ + 1) * K_;
  float* cV = Vc + (size_t)bh * (M_ + 1) * V_;

  if (tid < K_) qs[tid] = q_ws[(size_t)bh * K_ + tid];
  __syncthreads();

  const v4f qv = ((const v4f*)qs)[lane];

  // ---- logits[m] = q . K[m], with NT write-through copy prev_K -> Kc ----
  for (int m = wave; m <= M_; m += 8) {
    v4f kv;
    if (m < M_) {
      if (m + 8 < M_)
        __builtin_prefetch(pK + (size_t)(m + 8) * K_ + lane * 4, 0, 1);
      kv = __builtin_nontemporal_load((const v4f*)(pK + (size_t)m * K_) + lane);
      __builtin_nontemporal_store(kv, (v4f*)(cK + (size_t)m * K_) + lane);
    } else {
      kv = ((const v4f*)(cK + (size_t)M_ * K_))[lane]; // new row (kernel 1)
    }
    float p = kv.x * qv.x + kv.y * qv.y + kv.z * qv.z + kv.w * qv.w;
    #pragma unroll
    for (int off = 16; off > 0; off >>= 1) p += __shfl_xor(p, off, 32);
    if (lane == 0) logits[m] = p;
  }
  __syncthreads();

  // ---- softmax over M_+1 logits (max-subtracted, like jax.nn.softmax) ----
  float lm = -3.402823466e+38f;
  for (int i = tid; i <= M_; i += 256) lm = fmaxf(lm, logits[i]);
  red[tid] = lm; __syncthreads();
  for (int s = 128; s > 0; s >>= 1) {
    if (tid < s) red[tid] = fmaxf(red[tid], red[tid + s]);
    __syncthreads();
  }
  const float mx = red[0];
  __syncthreads();
  float ls = 0.f;
  for (int i = tid; i <= M_; i += 256) {
    const float e = __expf(logits[i] - mx);
    logits[i] = e;
    ls += e;
  }
  red[tid] = ls; __syncthreads();
  for (int s = 128; s > 0; s >>= 1) {
    if (tid < s) red[tid] += red[tid + s];
    __syncthreads();
  }
  if (tid == 0) s_inv = 1.f / red[0];
  __syncthreads();
  const float inv = s_inv;

  // ---- o[v] = sum_m w[m] * V[m,v], NT write-through copy prev_V -> Vc ----
  const int v    = tid & (V_ - 1);
  const int team = tid >> 7;               // 2 teams split m by parity
  float acc = 0.f;
  for (int m = team; m <= M_; m += 2) {
    float vv;
    if (m < M_) {
      vv = __builtin_nontemporal_load(pV + (size_t)m * V_ + v);
      __builtin_nontemporal_store(vv, cV + (size_t)m * V_ + v);
    } else {
      vv = cV[(size_t)M_ * V_ + v];        // new row (kernel 1)
    }
    acc += logits[m] * vv;
  }
  red[tid] = acc; __syncthreads();
  if (tid < V_) {
    o_ws[(size_t)bh * V_ + tid] = (red[tid] + red[tid + 128]) * inv;
  }
}

// ---------------------------------------------------------------------------
// Kernel 3: output projection y = o_flat[16 x 4096] @ Wo_flat[4096 x 4096]
// (h,v flattened: Wo is [H,V,D] so row f=h*V+v is contiguous at Wo + f*D).
// One block (4 waves) per 16-wide d tile; each wave takes a 1024-wide slice
// of the 4096 reduction via WMMA f32 16x16x4, partials combined in LDS.
// ---------------------------------------------------------------------------
__global__ __launch_bounds__(128)
void out_proj_wmma(const float* __restrict__ o_ws,
                   const float* __restrict__ Wo,
                   float* __restrict__ y) {
  __shared__ float part[4][32][8];

  const int lane = threadIdx.x & 31;
  const int wave = threadIdx.x >> 5;       // 0..3 : slice of the f reduction
  const int d0   = blockIdx.x * 16;
  const int n    = lane & 15;
  const int cof  = (lane < 16) ? 0 : 2;

  v8f acc = {};
  const int fbeg = wave * ((H_ * V_) / 4);
  const int fend = fbeg + ((H_ * V_) / 4);
  for (int f = fbeg; f < fend; f += 4) {
    const float* op = o_ws + (size_t)n * (H_ * V_) + f + cof;  // A: o[b=n][f..]
    v2f a; a.x = op[0]; a.y = op[1];
    const float* wp = Wo + (size_t)(f + cof) * D_ + d0 + n;    // B: Wo[f..][d0+n]
    v2f bt; bt.x = wp[0]; bt.y = wp[D_];
    acc = __builtin_amdgcn_wmma_f32_16x16x4_f32(false, a, false, bt,
                                                (short)0, acc, false, false);
  }

  if (wave > 0) {
    #pragma unroll
    for (int r = 0; r < 8; ++r) part[wave][lane][r] = acc[r];
  }
  __syncthreads();
  if (wave == 0) {
    const int mbase = (lane < 16) ? 0 : 8;
    #pragma unroll
    for (int r = 0; r < 8; ++r) {
      const float val = acc[r] + part[1][lane][r] + part[2][lane][r] + part[3][lane][r];
      y[(size_t)(mbase + r) * D_ + d0 + n] = val;
    }
  }
}

// ---------------------------------------------------------------------------
extern "C" void kernel_launch(void* const* d_in, const int* in_sizes, int n_in,
                              void* d_out, int out_size, void* d_ws, size_t ws_size,
                              hipStream_t stream) {
  const float* x     = (const float*)d_in[0];
  const float* prevK = (const float*)d_in[1];
  const float* prevV = (const float*)d_in[2];
  const float* Wq    = (const float*)d_in[3];
  const float* Wk    = (const float*)d_in[4];
  const float* Wv    = (const float*)d_in[5];
  const float* Wo    = (const float*)d_in[6];

  // d_out = [ y (B*D) | Kc (B*H*(M+1)*K) | Vc (B*H*(M+1)*V) ]
  float* y  = (float*)d_out;
  float* Kc = y  + (size_t)B_ * D_;
  float* Vc = Kc + (size_t)B_ * H_ * (M_ + 1) * K_;

  // workspace: q [B,H,K] then o [B,H,V]  (512 KB total)
  float* q_ws = (float*)d_ws;
  float* o_ws = q_ws + (size_t)B_ * H_ * K_;
  (void)in_sizes; (void)n_in; (void)out_size; (void)ws_size;

  qkv_proj_wmma<<<3 * H_ * 8, 128, 0, stream>>>(x, Wq, Wk, Wv, q_ws, Kc, Vc);
  attn_fused  <<<B_ * H_,    256, 0, stream>>>(prevK, prevV, q_ws, Kc, Vc, o_ws);
  out_proj_wmma<<<D_ / 16,   128, 0, stream>>>(o_ws, Wo, y);
}